// Audio_LSTM_39024072851972
// MI455X (gfx1250) — compile-verified
//
#include <hip/hip_runtime.h>
#include <hip/hip_bf16.h>

// 4-layer bidirectional LSTM (B=128, T=4096, H=13) for gfx1250.
//
// The T=4096 scan is the critical path: each (16-batch, direction) workgroup
// runs the whole time loop with weights in registers and state in LDS.
// Per-step matmul  g[16,52] = [h_t | x_t][16,K] * W^T[K,52] + b  (K = 13+Din)
// uses chained V_WMMA_F32_16X16X4_F32 (fp32-exact), one 16x16 N-tile per wave
// (waves 0..3 of an 8-wave block). Gate nonlinearities: v_exp_f32 + v_rcp_f32,
// one element per thread (208 of 256). Barriers are LDS-only split barriers
// (s_wait_dscnt + s_barrier_signal/-wait) so global h-stores never stall the
// scan. x(t+1) loads issue at the top of the gate phase and land behind it.

typedef __attribute__((ext_vector_type(2))) float v2f;
typedef __attribute__((ext_vector_type(8))) float v8f;

#define LSTM_H  13
#define LSTM_G4 52
#define LSTM_T  4096
#define LSTM_B  128

__device__ __forceinline__ float frcp(float x)     { return __builtin_amdgcn_rcpf(x); }
__device__ __forceinline__ float fsigmoid(float x) { return frcp(1.0f + __expf(-x)); }
__device__ __forceinline__ float ftanhf(float x)   { return 1.0f - 2.0f * frcp(1.0f + __expf(2.0f * x)); }

// Workgroup split barrier with LDS-only completion: does NOT wait on global
// store/load counters (unlike __syncthreads), keeping HBM latency off the scan.
__device__ __forceinline__ void wg_barrier_lds() {
    asm volatile(
        "s_wait_dscnt 0x0\n\t"
        "s_barrier_signal -1\n\t"
        "s_barrier_wait -1"
        ::: "memory");
}

template <int DIN>
__global__ __launch_bounds__(256) void lstm_layer_kernel(
    const float* __restrict__ x,    // [B, T, DIN]
    float* __restrict__ out,        // [B, T, 26]; this dir writes cols dir*13 .. dir*13+12
    const float* __restrict__ w_ih, // [2, 52, DIN]
    const float* __restrict__ w_hh, // [2, 52, 13]
    const float* __restrict__ b_ih, // [2, 52]
    const float* __restrict__ b_hh) // [2, 52]
{
    constexpr int H   = LSTM_H, G4 = LSTM_G4, T = LSTM_T;
    constexpr int KT  = H + DIN;              // live K (26 or 39)
    constexpr int KCH = (KT + 3) / 4;         // K=4 chunks (7 or 10)
    constexpr int SS  = 42;                   // state row stride (floats)
    constexpr int GS  = 60;                   // gate row stride, 8*60 % 64 != 0
    constexpr int NX  = (16 * DIN + 255) / 256; // x-load slots per thread (1 or 2)

    const int dir  = blockIdx.y;
    const int b0   = blockIdx.x * 16;
    const int tid  = threadIdx.x;
    const int lane = tid & 31;
    const int wv   = tid >> 5;                // wave id; waves 0..3 own N tiles
    const int half = lane >> 4;               // lane half selects K+2 / M+8
    const int l15  = lane & 15;

    __shared__ float S [16][SS];              // cols 0..12 = h_t, 13..13+DIN-1 = x_t, rest 0
    __shared__ float Wt[40][64];              // Wt[k][n] = W[n][k], zero padded
    __shared__ float Gm[16][GS];              // gate pre-activations [16,52]
    __shared__ float Cs[16][16];              // cell state [16,13]
    __shared__ float bias[64];

    const float* wih = w_ih + (size_t)dir * G4 * DIN;
    const float* whh = w_hh + (size_t)dir * G4 * H;

    // ---- stage W^T (K padded to 40, N padded to 64), bias, zero state ----
    for (int i = tid; i < 40 * 64; i += 256) {
        int k = i >> 6, nn = i & 63;
        float v = 0.0f;
        if (nn < G4) {
            if (k < H)       v = whh[nn * H + k];
            else if (k < KT) v = wih[nn * DIN + (k - H)];
        }
        Wt[k][nn] = v;
    }
    for (int i = tid; i < 64; i += 256)
        bias[i] = (i < G4) ? (b_ih[dir * G4 + i] + b_hh[dir * G4 + i]) : 0.0f;
    for (int i = tid; i < 16 * SS; i += 256) (&S[0][0])[i]  = 0.0f;
    if (tid < 256) (&Cs[0][0])[tid] = 0.0f;
    __syncthreads();

    // ---- per-wave register-resident B fragments + bias (used by waves 0..3) ----
    // B 4x16 layout: v.x -> K = 4c + 2*half, v.y -> K = 4c + 2*half + 1, col n
    const int n = (wv & 3) * 16 + l15;
    v2f bfrag[KCH];
    #pragma unroll
    for (int c = 0; c < KCH; ++c) {
        const int k = 4 * c + 2 * half;
        bfrag[c].x = Wt[k][n];
        bfrag[c].y = Wt[k + 1][n];
    }
    const float bn = bias[n];

    // ---- loop-carried 32-bit global offsets ----
    const int tt0   = dir ? (T - 1) : 0;
    const int xstep = dir ? -DIN : DIN;
    const int ostep = dir ? -26  : 26;

    // compact x-load slots: element i -> (row i/DIN, col i%DIN); only last slot predicated
    int    xoff[NX];
    float* sdst[NX];
    bool   xval[NX];
    #pragma unroll
    for (int s = 0; s < NX; ++s) {
        const int i  = tid + 256 * s;
        xval[s] = (i < 16 * DIN);
        const int ii = xval[s] ? i : 0;
        const int r = ii / DIN, c = ii % DIN;       // constant-divisor, setup-only
        xoff[s] = ((b0 + r) * T + tt0) * DIN + c;
        sdst[s] = &S[r][H + c];
    }
    // prefetch slot (threads 0..15 prefetch row tid's t+2 timestep)
    int poff = ((b0 + (tid & 15)) * T + tt0) * DIN + 2 * xstep;

    // gate slot: one element per thread, e -> (row e/13, unit e%13)
    const bool gv = (tid < 16 * H);                 // 208 active
    const int  ge = gv ? tid : 0;
    const int  gr = ge / H, gj = ge % H;            // constant-divisor, setup-only
    const float* gp = &Gm[gr][gj];                  // gates at +0, +13, +26, +39
    float*       cp = &Cs[gr][gj];
    float*       hp = &S[gr][gj];
    int ooff = ((b0 + gr) * T + tt0) * 26 + dir * H + gj;

    // ---- preload x(tt0) ----
    #pragma unroll
    for (int s = 0; s < NX; ++s)
        if (xval[s]) *sdst[s] = x[xoff[s]];
    __syncthreads();

    // ---- sequential scan: 2 LDS-only barriers per step ----
    for (int t = 0; t < T; ++t) {
        // phase 1 (waves 0..3): acc = bias + [h|x] * W^T, chained fp32 WMMA
        if (wv < 4) {
            v8f acc = {bn, bn, bn, bn, bn, bn, bn, bn};
            #pragma unroll
            for (int c = 0; c < KCH; ++c) {
                const int k = 4 * c + 2 * half;  // A 16x4: lane=row, half picks K pair
                v2f a;
                a.x = S[l15][k];
                a.y = S[l15][k + 1];
                acc = __builtin_amdgcn_wmma_f32_16x16x4_f32(
                    false, a, false, bfrag[c], (short)0, acc, false, false);
            }
            // D 16x16: component j -> row 8*half + j, col n
            if (n < G4) {
                #pragma unroll
                for (int j = 0; j < 8; ++j)
                    Gm[half * 8 + j][n] = acc[j];
            }
        }
        wg_barrier_lds();    // Gm ready; all S reads of this step complete

        // phase 2: issue x(t+1) loads first (latency hides behind gate math)
        const bool doload = (t + 1 < T);
        float xv[NX];
        if (doload) {
            #pragma unroll
            for (int s = 0; s < NX; ++s) {
                xoff[s] += xstep;
                if (xval[s]) xv[s] = x[xoff[s]];
            }
            if (t + 2 < T && tid < 16)
                __builtin_prefetch(&x[poff], 0, 1);   // global_prefetch_b8 for t+2
            poff += xstep;
        }

        // gate elementwise: one (row, unit) per thread
        if (gv) {
            const float gi = fsigmoid(gp[0]);
            const float gf = fsigmoid(gp[H]);
            const float gg = ftanhf (gp[2 * H]);
            const float go = fsigmoid(gp[3 * H]);
            const float cn = gf * (*cp) + gi * gg;
            *cp = cn;
            const float h = go * ftanhf(cn);
            *hp = h;                     // feeds next step's A
            out[ooff] = h;               // drains in background (no barrier wait)
            ooff += ostep;
        }

        // land x(t+1) into S x-columns
        if (doload) {
            #pragma unroll
            for (int s = 0; s < NX; ++s)
                if (xval[s]) *sdst[s] = xv[s];
        }
        wg_barrier_lds();    // S (h and x columns) ready for next wmma phase
    }
}

extern "C" void kernel_launch(void* const* d_in, const int* in_sizes, int n_in,
                              void* d_out, int out_size, void* d_ws, size_t ws_size,
                              hipStream_t stream) {
    (void)in_sizes; (void)n_in; (void)out_size; (void)ws_size;
    const float* x    = (const float*)d_in[0];
    const float* wih0 = (const float*)d_in[1];
    const float* whh0 = (const float*)d_in[2];
    const float* bih0 = (const float*)d_in[3];
    const float* bhh0 = (const float*)d_in[4];
    const float* wihR = (const float*)d_in[5];
    const float* whhR = (const float*)d_in[6];
    const float* bihR = (const float*)d_in[7];
    const float* bhhR = (const float*)d_in[8];
    float* out  = (float*)d_out;
    float* buf0 = (float*)d_ws;                           // [B,T,26]
    float* buf1 = buf0 + (size_t)LSTM_B * LSTM_T * 26;    // [B,T,26]

    dim3 grid(LSTM_B / 16, 2), blk(256);

    // layer 0: Din=13
    lstm_layer_kernel<13><<<grid, blk, 0, stream>>>(x, buf0, wih0, whh0, bih0, bhh0);

    // layers 1..3: Din=26, ping-pong buffers, last layer writes d_out
    const size_t WI = (size_t)2 * 52 * 26;
    const size_t WH = (size_t)2 * 52 * 13;
    const size_t BB = (size_t)2 * 52;
    float* ins [3] = {buf0, buf1, buf0};
    float* outs[3] = {buf1, buf0, out};
    for (int l = 0; l < 3; ++l) {
        lstm_layer_kernel<26><<<grid, blk, 0, stream>>>(
            ins[l], outs[l],
            wihR + (size_t)l * WI, whhR + (size_t)l * WH,
            bihR + (size_t)l * BB, bhhR + (size_t)l * BB);
    }
}